// AdaptiveMultiWIRE_36481452212323
// MI455X (gfx1250) — compile-verified
//
#include <hip/hip_runtime.h>

// AdaptiveMultiWIRE fused kernel for gfx1250 (MI455X).
// Layer-1 complex GEMMs run on V_WMMA_F32_16X16X4_F32 (exact fp32 tensor op).

typedef float v2f __attribute__((ext_vector_type(2)));
typedef float v8f __attribute__((ext_vector_type(8)));

#define INF   2
#define HID   181
#define OUTF  3
#define NCH   128
#define NSEL  64
#define NPTS  2048
#define RANK  4
#define HID2  (HID * HID)      // 32761

#define TM    128              // points per workgroup
#define KS    186              // padded K stride (even, 186%64=58 -> conflict-free 16-lane stride)
#define NT    12               // ceil(181/16) N tiles
#define KMAX  184              // 46 wmma k-steps of 4

// LDS layout (floats):
//   HR   [TM*KS]         layer-0 output real  (WMMA A, zero padded k>=181)
//   HI   [TM*KS]         layer-0 output imag
//   BT   [4][16*KS]      W panels col-major Bt[n][k]: {W1re, W1im, W2re, W2im}
//                        (reused as layer-0 weight staging before the N loop)
//   XT   [TM*2]          x tile
//   WFr  [192*3]         final weights real (zero padded)
//   WFi  [192*3]
#define SM_HR   0
#define SM_HI   (SM_HR + TM * KS)
#define SM_BT   (SM_HI + TM * KS)
#define SM_XT   (SM_BT + 4 * 16 * KS)
#define SM_WFR  (SM_XT + TM * 2)
#define SM_WFI  (SM_WFR + 192 * 3)
#define SM_TOT  (SM_WFI + 192 * 3)        // 60928 floats = 243712 bytes

__global__ __launch_bounds__(256) void wire_fused_kernel(
    const float* __restrict__ x,        const int* __restrict__ indices,
    const float* __restrict__ w0_lin,   const float* __restrict__ b0_lin,
    const float* __restrict__ w0_orth,  const float* __restrict__ b0_orth,
    const float* __restrict__ U1_lin,   const float* __restrict__ V1_lin,
    const float* __restrict__ b1_lin,   const float* __restrict__ U1_orth,
    const float* __restrict__ V1_orth,  const float* __restrict__ b1_orth,
    const float* __restrict__ Wf,       const float* __restrict__ bf,
    float* __restrict__ out)
{
    extern __shared__ float smem[];
    float* HR  = smem + SM_HR;
    float* HI_ = smem + SM_HI;
    float* BT  = smem + SM_BT;
    float* XT  = smem + SM_XT;
    float* WFr = smem + SM_WFR;
    float* WFi = smem + SM_WFI;

    const int tid  = threadIdx.x;
    const int lane = tid & 31;
    const int wave = tid >> 5;           // 0..7 -> M sub-tile
    const int mt   = blockIdx.x;         // 0..15
    const int s    = blockIdx.y;         // 0..63
    const int c    = indices[s];

    // rank-4 channel scales (per-thread registers)
    float u1r[RANK], u1i[RANK], u2r[RANK], u2i[RANK];
#pragma unroll
    for (int r = 0; r < RANK; ++r) {
        u1r[r] = U1_lin[(c * RANK + r) * 2];
        u1i[r] = U1_lin[(c * RANK + r) * 2 + 1];
        u2r[r] = U1_orth[(c * RANK + r) * 2];
        u2i[r] = U1_orth[(c * RANK + r) * 2 + 1];
    }

    // ---- stage x tile, final weights, layer-0 weights (into BT scratch) ----
    for (int i = tid; i < TM * 2; i += 256) {
        int m = i >> 1, d = i & 1;
        XT[i] = x[(s * NPTS + mt * TM + m) * INF + d];
    }
    for (int i = tid; i < 192 * OUTF; i += 256) {
        int n = i / OUTF, o = i - n * OUTF;
        float re = 0.f, im = 0.f;
        if (n < HID) {
            const float* p = Wf + ((c * HID + n) * OUTF + o) * 2;
            re = p[0]; im = p[1];
        }
        WFr[i] = re; WFi[i] = im;
    }
    float* W0S = BT;  // 1086 floats of layer-0 staging (BT reused later)
    for (int i = tid; i < 1086; i += 256) {
        float v;
        if      (i < 362)  v = w0_lin [c * 362 + i];
        else if (i < 724)  v = w0_orth[c * 362 + (i - 362)];
        else if (i < 905)  v = b0_lin [c * HID + (i - 724)];
        else               v = b0_orth[c * HID + (i - 905)];
        W0S[i] = v;
    }
    __syncthreads();

    // ---- layer 0: h = exp(i*30*lin - 100*(lin^2+lin2^2)), write A planes ----
    for (int i = tid; i < TM * KS; i += 256) {
        int m = i / KS, k = i - m * KS;
        float hr = 0.f, hi = 0.f;
        if (k < HID) {
            float x0 = XT[2 * m], x1 = XT[2 * m + 1];
            float lin  = x0 * W0S[k]       + x1 * W0S[HID + k]       + W0S[724 + k];
            float lin2 = x0 * W0S[362 + k] + x1 * W0S[362 + HID + k] + W0S[905 + k];
            float e = __expf(-100.f * (lin * lin + lin2 * lin2));
            float sn, cs; __sincosf(30.f * lin, &sn, &cs);
            hr = e * cs; hi = e * sn;
        }
        HR[i] = hr; HI_[i] = hi;
    }

    // per-lane partial accumulators of the final 181->3 complex contraction
    float oar[8][OUTF], oai[8][OUTF];
#pragma unroll
    for (int v = 0; v < 8; ++v)
#pragma unroll
        for (int o = 0; o < OUTF; ++o) { oar[v][o] = 0.f; oai[v][o] = 0.f; }

    const int nlane = lane & 15;
    const int khalf = 2 * (lane >> 4);                 // K pair select per WMMA layout
    const float* Ar  = HR  + (wave * 16 + nlane) * KS + khalf;
    const float* Ai  = HI_ + (wave * 16 + nlane) * KS + khalf;
    const float* B1r = BT + 0 * 16 * KS + nlane * KS + khalf;
    const float* B1i = BT + 1 * 16 * KS + nlane * KS + khalf;
    const float* B2r = BT + 2 * 16 * KS + nlane * KS + khalf;
    const float* B2i = BT + 3 * 16 * KS + nlane * KS + khalf;

    for (int nt = 0; nt < NT; ++nt) {
        __syncthreads();   // prior consumers of BT (and layer-0 staging) done
        // ---- expand rank-4 W panels: Bt[n][k] = sum_r U[c,r]*V[r, k*181+ng] ----
        for (int i = tid; i < 16 * KS; i += 256) {
            int n = i / KS, k = i - n * KS;
            int ng = nt * 16 + n;
            float b1re = 0.f, b1im = 0.f, b2re = 0.f, b2im = 0.f;
            if (k < HID && ng < HID) {
                int vidx = k * HID + ng;
#pragma unroll
                for (int r = 0; r < RANK; ++r) {
                    const float* v1 = V1_lin + (r * HID2 + vidx) * 2;
                    float vr = v1[0], vi = v1[1];
                    b1re += u1r[r] * vr - u1i[r] * vi;
                    b1im += u1r[r] * vi + u1i[r] * vr;
                    const float* v2 = V1_orth + (r * HID2 + vidx) * 2;
                    vr = v2[0]; vi = v2[1];
                    b2re += u2r[r] * vr - u2i[r] * vi;
                    b2im += u2r[r] * vi + u2i[r] * vr;
                }
            }
            BT[0 * 16 * KS + i] = b1re;
            BT[1 * 16 * KS + i] = b1im;
            BT[2 * 16 * KS + i] = b2re;
            BT[3 * 16 * KS + i] = b2im;
        }
        __syncthreads();

        // ---- complex GEMM on f32 WMMA: 4 real accumulators x 2 matrices ----
        v8f a1r = {0.f,0.f,0.f,0.f,0.f,0.f,0.f,0.f};
        v8f a1i = a1r, a2r = a1r, a2i = a1r;
        for (int kb = 0; kb < KMAX; kb += 4) {
            v2f ar  = *(const v2f*)(Ar  + kb);
            v2f ai  = *(const v2f*)(Ai  + kb);
            v2f ain = -ai;                              // F32 WMMA has no A/B neg
            v2f b1re = *(const v2f*)(B1r + kb);
            v2f b1im = *(const v2f*)(B1i + kb);
            v2f b2re = *(const v2f*)(B2r + kb);
            v2f b2im = *(const v2f*)(B2i + kb);
            a1r = __builtin_amdgcn_wmma_f32_16x16x4_f32(false, ar,  false, b1re, (short)0, a1r, false, false);
            a1r = __builtin_amdgcn_wmma_f32_16x16x4_f32(false, ain, false, b1im, (short)0, a1r, false, false);
            a1i = __builtin_amdgcn_wmma_f32_16x16x4_f32(false, ar,  false, b1im, (short)0, a1i, false, false);
            a1i = __builtin_amdgcn_wmma_f32_16x16x4_f32(false, ai,  false, b1re, (short)0, a1i, false, false);
            a2r = __builtin_amdgcn_wmma_f32_16x16x4_f32(false, ar,  false, b2re, (short)0, a2r, false, false);
            a2r = __builtin_amdgcn_wmma_f32_16x16x4_f32(false, ain, false, b2im, (short)0, a2r, false, false);
            a2i = __builtin_amdgcn_wmma_f32_16x16x4_f32(false, ar,  false, b2im, (short)0, a2i, false, false);
            a2i = __builtin_amdgcn_wmma_f32_16x16x4_f32(false, ai,  false, b2re, (short)0, a2i, false, false);
        }

        // ---- epilogue: bias + Gabor nonlinearity + fold into final contraction ----
        int ng = nt * 16 + nlane;
        float b1br = 0.f, b1bi = 0.f, b2br = 0.f, b2bi = 0.f;
        if (ng < HID) {
            const float* p = b1_lin + (c * HID + ng) * 2;
            b1br = p[0]; b1bi = p[1];
            p = b1_orth + (c * HID + ng) * 2;
            b2br = p[0]; b2bi = p[1];
        }
#pragma unroll
        for (int v = 0; v < 8; ++v) {
            float lr  = a1r[v] + b1br, li  = a1i[v] + b1bi;
            float l2r = a2r[v] + b2br, l2i = a2i[v] + b2bi;
            float e = __expf(-30.f * li - 100.f * (lr*lr + li*li + l2r*l2r + l2i*l2i));
            float sn, cs; __sincosf(30.f * lr, &sn, &cs);
            float h2r = e * cs, h2i = e * sn;
#pragma unroll
            for (int o = 0; o < OUTF; ++o) {
                float wr = WFr[ng * OUTF + o];          // zero-padded -> no NaN/garbage
                float wi = WFi[ng * OUTF + o];
                oar[v][o] += h2r * wr - h2i * wi;
                oai[v][o] += h2r * wi + h2i * wr;
            }
        }
    }

    // ---- reduce over the 16 N-lanes (C layout halves) and store complex64 ----
#pragma unroll
    for (int v = 0; v < 8; ++v) {
#pragma unroll
        for (int o = 0; o < OUTF; ++o) {
            float re = oar[v][o], im = oai[v][o];
#pragma unroll
            for (int msk = 1; msk < 16; msk <<= 1) {
                re += __shfl_xor(re, msk, 32);
                im += __shfl_xor(im, msk, 32);
            }
            if ((lane & 15) == 0) {
                int m = mt * TM + wave * 16 + v + 8 * (lane >> 4);
                float* op = out + ((s * NPTS + m) * OUTF + o) * 2;
                op[0] = re + bf[(c * OUTF + o) * 2];
                op[1] = im + bf[(c * OUTF + o) * 2 + 1];
            }
        }
    }
}

extern "C" void kernel_launch(void* const* d_in, const int* in_sizes, int n_in,
                              void* d_out, int out_size, void* d_ws, size_t ws_size,
                              hipStream_t stream) {
    (void)in_sizes; (void)n_in; (void)out_size; (void)d_ws; (void)ws_size;
    const float* x       = (const float*)d_in[0];
    const int*   indices = (const int*)  d_in[1];
    const float* w0_lin  = (const float*)d_in[2];
    const float* b0_lin  = (const float*)d_in[3];
    const float* w0_orth = (const float*)d_in[4];
    const float* b0_orth = (const float*)d_in[5];
    const float* U1_lin  = (const float*)d_in[6];
    const float* V1_lin  = (const float*)d_in[7];
    const float* b1_lin  = (const float*)d_in[8];
    const float* U1_orth = (const float*)d_in[9];
    const float* V1_orth = (const float*)d_in[10];
    const float* b1_orth = (const float*)d_in[11];
    const float* Wf      = (const float*)d_in[12];
    const float* bf      = (const float*)d_in[13];
    float*       out     = (float*)d_out;

    dim3 grid(NPTS / TM, NSEL);     // (16, 64)
    dim3 block(256);                // 8 wave32
    size_t smem = SM_TOT * sizeof(float);   // ~238 KB of the 320 KB WGP LDS
    wire_fused_kernel<<<grid, block, smem, stream>>>(
        x, indices, w0_lin, b0_lin, w0_orth, b0_orth,
        U1_lin, V1_lin, b1_lin, U1_orth, V1_orth, b1_orth, Wf, bf, out);
}